// Gma3D_71305047049031
// MI455X (gfx1250) — compile-verified
//
#include <hip/hip_runtime.h>
#include <hip/hip_bf16.h>

// ---------------------------------------------------------------------------
// GMA attention block for MI455X (gfx1250, wave32).
// Flash-style fusion: 4096x4096 attention matrices never materialized; 16x16
// corr tiles recomputed with v_wmma_f32_16x16x4_f32 (K=16 -> 4 WMMA) in each
// pass. Data layouts are packed so every WMMA operand / stats fetch is a
// b128/b64 vector load. Final pass feeds attn tiles (LDS-staged) into a
// second WMMA chain for gma = xv @ attn.
// ---------------------------------------------------------------------------

typedef __attribute__((ext_vector_type(2))) float v2f;
typedef __attribute__((ext_vector_type(8))) float v8f;

#define NB 2
#define NC 64
#define NQ 16
#define NN 4096
#define NG 8
#define EPS 1e-9f
#define POSC 0.1f
#define GN_EPS 1e-5f

__device__ __forceinline__ v8f wmma4(v2f a, v2f b, v8f c) {
  // D = A(16x4,f32) x B(4x16,f32) + C(16x16,f32)
  return __builtin_amdgcn_wmma_f32_16x16x4_f32(false, a, false, b, (short)0, c,
                                               false, false);
}

// q is stored permuted-transposed: qr[n][j], j = half*8 + 2*g + e for channel
// k = 4*g + 2*half + e.  A lane (l16, half) then gets all 4 WMMA k-chunk
// operands from TWO contiguous b128 loads (ISA 7.12.2 fp32 A/B striping:
// VGPR0 = A[l%16, half?2:0], VGPR1 = A[l%16, half?3:1]).
__device__ __forceinline__ void load_qk4(const float* __restrict__ qrb, int base,
                                         int half, int l16, v2f ops[4]) {
  const float4* p = (const float4*)(qrb + (base + l16) * 16 + half * 8);
  float4 lo = p[0], hi = p[1];
  ops[0].x = lo.x; ops[0].y = lo.y;
  ops[1].x = lo.z; ops[1].y = lo.w;
  ops[2].x = hi.x; ops[2].y = hi.y;
  ops[3].x = hi.z; ops[3].y = hi.w;
}

__device__ __forceinline__ void olcomb(float& m, float& s, float m2, float s2) {
  float mn = fmaxf(m, m2);
  s = s * __expf(m - mn) + s2 * __expf(m2 - mn);
  m = mn;
}

// ---------------------------------------------------------------------------
// K0: qr = permuted(Wqk@ctx), xv = Wv@motion + bv, pt = {x,y,z,|p|^2}
// ---------------------------------------------------------------------------
__global__ __launch_bounds__(256) void k_qxv(
    const float* __restrict__ ctx, const float* __restrict__ mot,
    const float* __restrict__ xyz, const float* __restrict__ Wqk,
    const float* __restrict__ Wv, const float* __restrict__ bv,
    float* __restrict__ qr, float* __restrict__ xv, float4* __restrict__ pt) {
  __shared__ float sWqk[NQ * NC];
  __shared__ float sWv[NC * NC];
  __shared__ float sbv[NC];
  int tid = threadIdx.x;
  for (int i = tid; i < NQ * NC; i += 256) sWqk[i] = Wqk[i];
  for (int i = tid; i < NC * NC; i += 256) sWv[i] = Wv[i];
  if (tid < NC) sbv[tid] = bv[tid];
  __syncthreads();

  int b = blockIdx.x;
  int n = blockIdx.y * 256 + tid;
  float col[NC];
#pragma unroll
  for (int c = 0; c < NC; ++c) col[c] = ctx[b * NC * NN + c * NN + n];
  float qv[16];
#pragma unroll
  for (int o = 0; o < NQ; ++o) {
    float s = 0.f;
#pragma unroll
    for (int c = 0; c < NC; ++c) s += sWqk[o * NC + c] * col[c];
    int j = (((o >> 1) & 1) << 3) + ((o >> 2) << 1) + (o & 1);
    qv[j] = s;
  }
  float4* qdst = (float4*)(qr + (b * NN + n) * 16);
#pragma unroll
  for (int i = 0; i < 4; ++i)
    qdst[i] = make_float4(qv[4 * i], qv[4 * i + 1], qv[4 * i + 2], qv[4 * i + 3]);

#pragma unroll
  for (int c = 0; c < NC; ++c) col[c] = mot[b * NC * NN + c * NN + n];
  for (int o = 0; o < NC; ++o) {
    float s = sbv[o];
#pragma unroll
    for (int c = 0; c < NC; ++c) s += sWv[o * NC + c] * col[c];
    xv[b * NC * NN + o * NN + n] = s;
  }
  float x = xyz[b * NN * 3 + n * 3 + 0];
  float y = xyz[b * NN * 3 + n * 3 + 1];
  float z = xyz[b * NN * 3 + n * 3 + 2];
  pt[b * NN + n] = make_float4(x, y, z, x * x + y * y + z * z);
}

// ---------------------------------------------------------------------------
// K1: per-row softmax stats of corr -> rstat[m].x = rowmax, .y = 1/rowsum
// ---------------------------------------------------------------------------
__global__ __launch_bounds__(256) void k_rowstats(const float* __restrict__ qr,
                                                  float* __restrict__ rstat) {
  int b = blockIdx.x, m0 = blockIdx.y * 16;
  int tid = threadIdx.x, wave = tid >> 5, lane = tid & 31;
  int half = lane >> 4, l16 = lane & 15;
  const float* qrb = qr + b * NN * 16;

  v2f A[4];
  load_qk4(qrb, m0, half, l16, A);
  float rm[8], rs[8];
#pragma unroll
  for (int v = 0; v < 8; ++v) { rm[v] = -1e30f; rs[v] = 0.f; }

  for (int nt = wave; nt < NN / 16; nt += 8) {
    if (nt + 8 < NN / 16)
      __builtin_prefetch(qrb + ((nt + 8) * 16 + l16) * 16 + half * 8, 0, 0);
    v2f Bv[4];
    load_qk4(qrb, nt * 16, half, l16, Bv);
    v8f c = {};
#pragma unroll
    for (int kb = 0; kb < 4; ++kb) c = wmma4(A[kb], Bv[kb], c);
#pragma unroll
    for (int v = 0; v < 8; ++v) {
      float x = c[v];
      float mn = fmaxf(rm[v], x);
      rs[v] = rs[v] * __expf(rm[v] - mn) + __expf(x - mn);
      rm[v] = mn;
    }
  }
#pragma unroll
  for (int off = 8; off >= 1; off >>= 1) {
#pragma unroll
    for (int v = 0; v < 8; ++v) {
      float m2 = __shfl_xor(rm[v], off, 32);
      float s2 = __shfl_xor(rs[v], off, 32);
      olcomb(rm[v], rs[v], m2, s2);
    }
  }
  __shared__ float smax[8][16], ssum[8][16];
  if (l16 == 0) {
#pragma unroll
    for (int v = 0; v < 8; ++v) {
      smax[wave][v + 8 * half] = rm[v];
      ssum[wave][v + 8 * half] = rs[v];
    }
  }
  __syncthreads();
  if (tid < 16) {
    float m = smax[0][tid], s = ssum[0][tid];
    for (int w = 1; w < 8; ++w) olcomb(m, s, smax[w][tid], ssum[w][tid]);
    *(float2*)(rstat + (b * NN + m0 + tid) * 4) = make_float2(m, 1.0f / s);
  }
}

// ---------------------------------------------------------------------------
// K2: S[n] = sum_m softmax_row(corr)[m,n] -> invS = 1/(1e-9+S)
// ---------------------------------------------------------------------------
__global__ __launch_bounds__(256) void k_colsumS(const float* __restrict__ qr,
                                                 const float* __restrict__ rstat,
                                                 float* __restrict__ invS) {
  int b = blockIdx.x, n0 = blockIdx.y * 16;
  int tid = threadIdx.x, wave = tid >> 5, lane = tid & 31;
  int half = lane >> 4, l16 = lane & 15;
  const float* qrb = qr + b * NN * 16;

  v2f Bv[4];
  load_qk4(qrb, n0, half, l16, Bv);
  float acc = 0.f;
  for (int mt = wave; mt < NN / 16; mt += 8) {
    int m0 = mt * 16;
    if (mt + 8 < NN / 16)
      __builtin_prefetch(qrb + ((mt + 8) * 16 + l16) * 16 + half * 8, 0, 0);
    v2f A[4];
    load_qk4(qrb, m0, half, l16, A);
    v8f c = {};
#pragma unroll
    for (int kb = 0; kb < 4; ++kb) c = wmma4(A[kb], Bv[kb], c);
    int r0 = m0 + 8 * half;
#pragma unroll
    for (int v = 0; v < 8; ++v) {
      float2 st = *(const float2*)(rstat + (b * NN + r0 + v) * 4);
      acc += __expf(c[v] - st.x) * st.y;
    }
  }
  acc += __shfl_xor(acc, 16, 32);  // combine halves (same column)
  __shared__ float sacc[8][16];
  if (half == 0) sacc[wave][l16] = acc;
  __syncthreads();
  if (tid < 16) {
    float s = 0.f;
    for (int w = 0; w < 8; ++w) s += sacc[w][tid];
    invS[b * NN + n0 + tid] = 1.0f / (EPS + s);
  }
}

// ---------------------------------------------------------------------------
// K3: Dsum[n] = sum_m max(d,0) -> invDsum = 1/(1e-9+Dsum)
// ---------------------------------------------------------------------------
__global__ __launch_bounds__(256) void k_dsum(const float4* __restrict__ pt,
                                              float* __restrict__ invDsum) {
  int b = blockIdx.x;
  int n = blockIdx.y * 256 + threadIdx.x;
  float4 pn = pt[b * NN + n];
  float acc = 0.f;
  for (int m = 0; m < NN; ++m) {
    float4 pm = pt[b * NN + m];
    float d = pm.w + pn.w - 2.f * (pm.x * pn.x + pm.y * pn.y + pm.z * pn.z);
    acc += fmaxf(d, 0.f);
  }
  invDsum[b * NN + n] = 1.0f / (EPS + acc);
}

// ---------------------------------------------------------------------------
// K4: per-row softmax stats of local = attn1*mask -> rstat[m].z/.w
// ---------------------------------------------------------------------------
__global__ __launch_bounds__(256) void k_localrow(
    const float* __restrict__ qr, const float4* __restrict__ pt,
    const float* __restrict__ rstat, const float* __restrict__ invS,
    const float* __restrict__ invDsum, float* __restrict__ rstat_out) {
  int b = blockIdx.x, m0 = blockIdx.y * 16;
  int tid = threadIdx.x, wave = tid >> 5, lane = tid & 31;
  int half = lane >> 4, l16 = lane & 15;
  const float* qrb = qr + b * NN * 16;

  v2f A[4];
  load_qk4(qrb, m0, half, l16, A);
  int r0 = m0 + 8 * half;
  float4 ptm[8];
  float rmv[8], irs[8];
#pragma unroll
  for (int v = 0; v < 8; ++v) {
    ptm[v] = pt[b * NN + r0 + v];
    float2 st = *(const float2*)(rstat + (b * NN + r0 + v) * 4);
    rmv[v] = st.x;
    irs[v] = st.y;
  }
  float lm[8], ls[8];
#pragma unroll
  for (int v = 0; v < 8; ++v) { lm[v] = -1e30f; ls[v] = 0.f; }

  for (int nt = wave; nt < NN / 16; nt += 8) {
    int n0 = nt * 16, n = n0 + l16;
    if (nt + 8 < NN / 16)
      __builtin_prefetch(qrb + ((nt + 8) * 16 + l16) * 16 + half * 8, 0, 0);
    v2f Bv[4];
    load_qk4(qrb, n0, half, l16, Bv);
    v8f c = {};
#pragma unroll
    for (int kb = 0; kb < 4; ++kb) c = wmma4(A[kb], Bv[kb], c);
    float iS = invS[b * NN + n], iD = invDsum[b * NN + n];
    float4 pn = pt[b * NN + n];
#pragma unroll
    for (int v = 0; v < 8; ++v) {
      float a1 = __expf(c[v] - rmv[v]) * irs[v] * iS;
      float d = fmaxf(ptm[v].w + pn.w -
                          2.f * (ptm[v].x * pn.x + ptm[v].y * pn.y + ptm[v].z * pn.z),
                      0.f);
      float loc = (d * iD <= POSC) ? a1 : 0.f;
      float mn = fmaxf(lm[v], loc);
      ls[v] = ls[v] * __expf(lm[v] - mn) + __expf(loc - mn);
      lm[v] = mn;
    }
  }
#pragma unroll
  for (int off = 8; off >= 1; off >>= 1) {
#pragma unroll
    for (int v = 0; v < 8; ++v) {
      float m2 = __shfl_xor(lm[v], off, 32);
      float s2 = __shfl_xor(ls[v], off, 32);
      olcomb(lm[v], ls[v], m2, s2);
    }
  }
  __shared__ float smax[8][16], ssum[8][16];
  if (l16 == 0) {
#pragma unroll
    for (int v = 0; v < 8; ++v) {
      smax[wave][v + 8 * half] = lm[v];
      ssum[wave][v + 8 * half] = ls[v];
    }
  }
  __syncthreads();
  if (tid < 16) {
    float m = smax[0][tid], s = ssum[0][tid];
    for (int w = 1; w < 8; ++w) olcomb(m, s, smax[w][tid], ssum[w][tid]);
    *(float2*)(rstat_out + (b * NN + m0 + tid) * 4 + 2) = make_float2(m, 1.0f / s);
  }
}

// ---------------------------------------------------------------------------
// K5: T[n] = sum_m softmax_row(local)[m,n] -> invT = 1/(1e-9+T)
// ---------------------------------------------------------------------------
__global__ __launch_bounds__(256) void k_colsumT(
    const float* __restrict__ qr, const float4* __restrict__ pt,
    const float* __restrict__ rstat, const float* __restrict__ invS,
    const float* __restrict__ invDsum, float* __restrict__ invT) {
  int b = blockIdx.x, n0 = blockIdx.y * 16;
  int tid = threadIdx.x, wave = tid >> 5, lane = tid & 31;
  int half = lane >> 4, l16 = lane & 15;
  const float* qrb = qr + b * NN * 16;
  const float4* rst4 = (const float4*)rstat;

  v2f Bv[4];
  load_qk4(qrb, n0, half, l16, Bv);
  int n = n0 + l16;
  float iS = invS[b * NN + n], iD = invDsum[b * NN + n];
  float4 pn = pt[b * NN + n];

  float acc = 0.f;
  for (int mt = wave; mt < NN / 16; mt += 8) {
    int m0 = mt * 16;
    if (mt + 8 < NN / 16)
      __builtin_prefetch(qrb + ((mt + 8) * 16 + l16) * 16 + half * 8, 0, 0);
    v2f A[4];
    load_qk4(qrb, m0, half, l16, A);
    v8f c = {};
#pragma unroll
    for (int kb = 0; kb < 4; ++kb) c = wmma4(A[kb], Bv[kb], c);
    int r0 = m0 + 8 * half;
#pragma unroll
    for (int v = 0; v < 8; ++v) {
      float4 st = rst4[b * NN + r0 + v];  // {rowmax, invrowsum, lrowmax, invlrowsum}
      float4 pm = pt[b * NN + r0 + v];
      float a1 = __expf(c[v] - st.x) * st.y * iS;
      float d = fmaxf(pm.w + pn.w - 2.f * (pm.x * pn.x + pm.y * pn.y + pm.z * pn.z), 0.f);
      float loc = (d * iD <= POSC) ? a1 : 0.f;
      acc += __expf(loc - st.z) * st.w;
    }
  }
  acc += __shfl_xor(acc, 16, 32);
  __shared__ float sacc[8][16];
  if (half == 0) sacc[wave][l16] = acc;
  __syncthreads();
  if (tid < 16) {
    float s = 0.f;
    for (int w = 0; w < 8; ++w) s += sacc[w][tid];
    invT[b * NN + n0 + tid] = 1.0f / (EPS + s);
  }
}

// ---------------------------------------------------------------------------
// K6: gma[c,n] = sum_m xv[c,m]*(attn1+local2)[m,n] via second WMMA chain,
//     then res = Wt @ (motion - gma) + bt within the block's n-tile.
// ---------------------------------------------------------------------------
__global__ __launch_bounds__(256) void k_gma(
    const float* __restrict__ qr, const float* __restrict__ xv,
    const float* __restrict__ mot, const float4* __restrict__ pt,
    const float* __restrict__ rstat, const float* __restrict__ invS,
    const float* __restrict__ invDsum, const float* __restrict__ invT,
    const float* __restrict__ Wt, const float* __restrict__ bt,
    float* __restrict__ res) {
  __shared__ float sWt[NC * NC];   // 16 KB
  __shared__ float sbt[NC];
  __shared__ float sA[8][16][16];  // per-wave attn tile: C-layout -> B-layout
  __shared__ float sgma[NC][16];

  int b = blockIdx.x, n0 = blockIdx.y * 16;
  int tid = threadIdx.x, wave = tid >> 5, lane = tid & 31;
  int half = lane >> 4, l16 = lane & 15;
  const float* qrb = qr + b * NN * 16;
  const float* xvb = xv + b * NC * NN;
  const float4* rst4 = (const float4*)rstat;

  for (int i = tid; i < NC * NC; i += 256) sWt[i] = Wt[i];
  if (tid < NC) sbt[tid] = bt[tid];
  for (int i = tid; i < NC * 16; i += 256) ((float*)sgma)[i] = 0.f;
  __syncthreads();

  v2f Bq[4];
  load_qk4(qrb, n0, half, l16, Bq);
  int n = n0 + l16;
  float iS = invS[b * NN + n], iD = invDsum[b * NN + n], iT = invT[b * NN + n];
  float4 pn = pt[b * NN + n];

  v8f acc[4] = {v8f{}, v8f{}, v8f{}, v8f{}};

  for (int mt = wave; mt < NN / 16; mt += 8) {
    int m0 = mt * 16;
    if (mt + 8 < NN / 16)
      __builtin_prefetch(qrb + ((mt + 8) * 16 + l16) * 16 + half * 8, 0, 0);
    v2f A[4];
    load_qk4(qrb, m0, half, l16, A);
    v8f c = {};
#pragma unroll
    for (int kb = 0; kb < 4; ++kb) c = wmma4(A[kb], Bq[kb], c);

    int r0 = m0 + 8 * half;
#pragma unroll
    for (int v = 0; v < 8; ++v) {
      float4 st = rst4[b * NN + r0 + v];
      float4 pm = pt[b * NN + r0 + v];
      float a1 = __expf(c[v] - st.x) * st.y * iS;
      float d = fmaxf(pm.w + pn.w - 2.f * (pm.x * pn.x + pm.y * pn.y + pm.z * pn.z), 0.f);
      float loc = (d * iD <= POSC) ? a1 : 0.f;
      float l2 = __expf(loc - st.z) * st.w * iT;
      sA[wave][v + 8 * half][l16] = a1 + l2;
    }
    // gma accumulation: acc[ct] += xv[16ct.., m0+4kb..] * A[4kb.., n-tile]
#pragma unroll
    for (int kb = 0; kb < 4; ++kb) {
      v2f bb;
      bb.x = sA[wave][4 * kb + 2 * half + 0][l16];
      bb.y = sA[wave][4 * kb + 2 * half + 1][l16];
#pragma unroll
      for (int ct = 0; ct < 4; ++ct) {
        float2 av = *(const float2*)(xvb + (16 * ct + l16) * NN + m0 + 4 * kb + 2 * half);
        v2f aa;
        aa.x = av.x;
        aa.y = av.y;
        acc[ct] = wmma4(aa, bb, acc[ct]);
      }
    }
  }
  // cross-wave reduce gma into LDS
#pragma unroll
  for (int ct = 0; ct < 4; ++ct)
#pragma unroll
    for (int v = 0; v < 8; ++v)
      atomicAdd(&sgma[16 * ct + v + 8 * half][l16], acc[ct][v]);
  __syncthreads();

  // res[o, n0+nl] = sum_c Wt[o,c]*(motion[c,n]-gma[c,n]) + bt[o]
  int o = tid & 63;
  int g0 = tid >> 6;  // 0..3
#pragma unroll
  for (int r = 0; r < 4; ++r) {
    int nl = g0 + 4 * r;
    float s = sbt[o];
    for (int cc = 0; cc < NC; ++cc)
      s += sWt[o * NC + cc] * (mot[b * NC * NN + cc * NN + n0 + nl] - sgma[cc][nl]);
    res[b * NC * NN + o * NN + n0 + nl] = s;
  }
}

// ---------------------------------------------------------------------------
// K7: GroupNorm stats per (b, g)
// ---------------------------------------------------------------------------
__global__ __launch_bounds__(256) void k_gstats(const float* __restrict__ res,
                                                float* __restrict__ mu,
                                                float* __restrict__ rsig) {
  int b = blockIdx.x / NG, g = blockIdx.x % NG;
  int tid = threadIdx.x;
  float s = 0.f, s2 = 0.f;
  for (int i = tid; i < 8 * NN; i += 256) {
    int c = g * 8 + (i >> 12);
    int n = i & (NN - 1);
    float v = res[b * NC * NN + c * NN + n];
    s += v;
    s2 += v * v;
  }
  __shared__ float rs[256], rq[256];
  rs[tid] = s; rq[tid] = s2;
  __syncthreads();
  for (int off = 128; off >= 1; off >>= 1) {
    if (tid < off) { rs[tid] += rs[tid + off]; rq[tid] += rq[tid + off]; }
    __syncthreads();
  }
  if (tid == 0) {
    float inv = 1.0f / (8.0f * NN);
    float m = rs[0] * inv;
    float var = rq[0] * inv - m * m;
    mu[b * NG + g] = m;
    rsig[b * NG + g] = rsqrtf(var + GN_EPS);
  }
}

// ---------------------------------------------------------------------------
// K8: out = alpha * PReLU(GN(res)) + motion
// ---------------------------------------------------------------------------
__global__ __launch_bounds__(256) void k_final(
    const float* __restrict__ res, const float* __restrict__ mot,
    const float* __restrict__ mu, const float* __restrict__ rsig,
    const float* __restrict__ gn_w, const float* __restrict__ gn_b,
    const float* __restrict__ prelu_a, const float* __restrict__ alpha,
    float* __restrict__ out) {
  int idx = blockIdx.x * 256 + threadIdx.x;
  int b = idx / (NC * NN);
  int c = (idx / NN) % NC;
  int g = c >> 3;  // NC/NG == 8
  float x = res[idx];
  float gn = (x - mu[b * NG + g]) * rsig[b * NG + g] * gn_w[c] + gn_b[c];
  float pa = prelu_a[0];
  float pr = gn >= 0.f ? gn : pa * gn;
  out[idx] = alpha[0] * pr + mot[idx];
}

// ---------------------------------------------------------------------------
extern "C" void kernel_launch(void* const* d_in, const int* in_sizes, int n_in,
                              void* d_out, int out_size, void* d_ws, size_t ws_size,
                              hipStream_t stream) {
  const float* ctx   = (const float*)d_in[0];
  const float* mot   = (const float*)d_in[1];
  const float* xyz   = (const float*)d_in[2];
  const float* Wqk   = (const float*)d_in[3];
  const float* Wv    = (const float*)d_in[4];
  const float* bv    = (const float*)d_in[5];
  const float* Wt    = (const float*)d_in[6];
  const float* bt    = (const float*)d_in[7];
  const float* gn_w  = (const float*)d_in[8];
  const float* gn_b  = (const float*)d_in[9];
  const float* pre_a = (const float*)d_in[10];
  const float* alpha = (const float*)d_in[11];
  float* out = (float*)d_out;

  float* w = (float*)d_ws;
  float* qr      = w; w += NB * NN * 16;       // permuted q^T
  float* xv      = w; w += NB * NC * NN;
  float* ptf     = w; w += NB * NN * 4;        // {x,y,z,|p|^2}
  float* rstat   = w; w += NB * NN * 4;        // {rowmax, invrowsum, lrowmax, invlrowsum}
  float* invS    = w; w += NB * NN;
  float* invDsum = w; w += NB * NN;
  float* invT    = w; w += NB * NN;
  float* res     = w; w += NB * NC * NN;
  float* mu      = w; w += NB * NG;
  float* rsig    = w; w += NB * NG;
  float4* pt = (float4*)ptf;

  dim3 blk(256);
  k_qxv<<<dim3(NB, NN / 256), blk, 0, stream>>>(ctx, mot, xyz, Wqk, Wv, bv, qr, xv, pt);
  k_rowstats<<<dim3(NB, NN / 16), blk, 0, stream>>>(qr, rstat);
  k_dsum<<<dim3(NB, NN / 256), blk, 0, stream>>>(pt, invDsum);
  k_colsumS<<<dim3(NB, NN / 16), blk, 0, stream>>>(qr, rstat, invS);
  k_localrow<<<dim3(NB, NN / 16), blk, 0, stream>>>(qr, pt, rstat, invS, invDsum, rstat);
  k_colsumT<<<dim3(NB, NN / 16), blk, 0, stream>>>(qr, pt, rstat, invS, invDsum, invT);
  k_gma<<<dim3(NB, NN / 16), blk, 0, stream>>>(qr, xv, mot, pt, rstat, invS, invDsum,
                                               invT, Wt, bt, res);
  k_gstats<<<NB * NG, blk, 0, stream>>>(res, mu, rsig);
  k_final<<<(NB * NC * NN) / 256, blk, 0, stream>>>(res, mot, mu, rsig, gn_w, gn_b,
                                                    pre_a, alpha, out);
}